// LinearEncoder_45148696215964
// MI455X (gfx1250) — compile-verified
//
#include <hip/hip_runtime.h>
#include <hip/hip_bf16.h>

typedef float v2f __attribute__((ext_vector_type(2)));
typedef float v8f __attribute__((ext_vector_type(8)));

#define LDS_LD 130   // 16-row LDS tile stride (floats); breaks 64-bank conflicts

// ---------------------------------------------------------------------------
// WMMA f32 16x16x4 wrapper: D = A(16x4) * B(4x16) + C
// ---------------------------------------------------------------------------
__device__ __forceinline__ v8f wmma_f32(v2f a, v2f b, v8f c) {
    return __builtin_amdgcn_wmma_f32_16x16x4_f32(
        /*neg_a=*/false, a, /*neg_b=*/false, b,
        /*c_mod=*/(short)0, c, /*reuse_a=*/false, /*reuse_b=*/false);
}

// A fragment (16x4): lanes 0-15 -> M=lane, K={k,k+1}; lanes 16-31 -> K={k+2,k+3}
__device__ __forceinline__ v2f a_frag_g(const float* __restrict__ A, int lda,
                                        int r0, int k, int lane) {
    int m  = lane & 15;
    int kk = k + ((lane >> 4) << 1);
    const float* p = A + (size_t)(r0 + m) * lda + kk;
    return v2f{p[0], p[1]};
}

__device__ __forceinline__ v2f a_frag_lds(const float* __restrict__ L, int ld,
                                          int k, int lane) {
    int m  = lane & 15;
    int kk = k + ((lane >> 4) << 1);
    const float* p = L + m * ld + kk;
    return v2f{p[0], p[1]};
}

// B fragment (4x16): lanes 0-15 -> rows {k,k+1}; lanes 16-31 -> rows {k+2,k+3}
__device__ __forceinline__ v2f b_frag_g(const float* __restrict__ B, int ldb,
                                        int k, int c0, int lane) {
    int n  = lane & 15;
    int kk = k + ((lane >> 4) << 1);
    const float* p = B + (size_t)kk * ldb + c0 + n;
    return v2f{p[0], p[ldb]};
}

// C/D tile (16x16 f32): VGPR r holds row r (lanes 0-15) / row r+8 (lanes 16-31)
__device__ __forceinline__ void store_tile(float* __restrict__ D, int ldd,
                                           int r0, int c0, v8f acc, int lane,
                                           float bias) {
    int n    = lane & 15;
    int half = lane >> 4;
#pragma unroll
    for (int r = 0; r < 8; ++r)
        D[(size_t)(r0 + r + 8 * half) * ldd + c0 + n] = acc[r] + bias;
}

// ---------------------------------------------------------------------------
// Degree / normalization
// ---------------------------------------------------------------------------
__global__ void deg_init_kernel(float* __restrict__ deg, int N) {
    int i = blockIdx.x * blockDim.x + threadIdx.x;
    if (i < N) deg[i] = 1.0f;   // self-loop contribution
}

__global__ void deg_acc_kernel(const int* __restrict__ col, float* __restrict__ deg, int E) {
    int e = blockIdx.x * blockDim.x + threadIdx.x;
    if (e < E) atomicAdd(&deg[col[e]], 1.0f);
}

__global__ void dis_kernel(const float* __restrict__ deg, float* __restrict__ dis, int N) {
    int i = blockIdx.x * blockDim.x + threadIdx.x;
    if (i < N) dis[i] = rsqrtf(deg[i]);   // deg >= 1 always (self-loops)
}

// ---------------------------------------------------------------------------
// Fused self-branch + gcn1-transform kernel.
// One wave per 16-row strip. A fragments of x are register-cached once
// (ax[32] = 64 VGPRs) and reused across all 20 output-column tiles; the
// relu intermediate is staged in LDS and its A fragments cached in al[32].
// Requires N % 64 == 0 (true: 40000).
// ---------------------------------------------------------------------------
__global__ __launch_bounds__(128) void self_branch_kernel(
    const float* __restrict__ x, const float* __restrict__ Win,
    const float* __restrict__ bin, const float* __restrict__ Wout,
    const float* __restrict__ bout, const float* __restrict__ Wg1,
    float* __restrict__ l1_out, float* __restrict__ h1_out, int N)
{
    __shared__ float lds[4 * 16 * LDS_LD];
    const int lane = threadIdx.x & 31;
    const int wave = threadIdx.x >> 5;
    const int r0   = (blockIdx.x * 4 + wave) * 16;
    const int n    = lane & 15, half = lane >> 4;
    float* myL = lds + wave * 16 * LDS_LD;
    if (r0 >= N) return;   // never taken for N=40000; barrier stays uniform

    // Register-cache the x A-operand for this strip (K = 128 -> 32 fragments)
    v2f ax[32];
#pragma unroll
    for (int kc = 0; kc < 32; ++kc)
        ax[kc] = a_frag_g(x, 128, r0, kc * 4, lane);

    // Stage 1: l1a = relu(x @ Win + bin) -> LDS (8 col tiles of 16)
    for (int t = 0; t < 8; ++t) {
        v8f acc = {};
#pragma unroll
        for (int kc = 0; kc < 32; ++kc)
            acc = wmma_f32(ax[kc], b_frag_g(Win, 128, kc * 4, t * 16, lane), acc);
        float bb = bin[t * 16 + n];
#pragma unroll
        for (int r = 0; r < 8; ++r) {
            float v = acc[r] + bb;
            myL[(r + 8 * half) * LDS_LD + t * 16 + n] = v > 0.f ? v : 0.f;
        }
    }
    __syncthreads();   // all waves uniform; orders LDS writes before reads

    // Stage 1b: h1 = x @ Wg1 (bias applied after aggregation) — reuses ax
    for (int t = 0; t < 8; ++t) {
        v8f acc = {};
#pragma unroll
        for (int kc = 0; kc < 32; ++kc)
            acc = wmma_f32(ax[kc], b_frag_g(Wg1, 128, kc * 4, t * 16, lane), acc);
        store_tile(h1_out, 128, r0, t * 16, acc, lane, 0.f);
    }

    // Register-cache the l1a A-operand from LDS
    v2f al[32];
#pragma unroll
    for (int kc = 0; kc < 32; ++kc)
        al[kc] = a_frag_lds(myL, LDS_LD, kc * 4, lane);

    // Stage 2: l1 = x @ Wout[0:128] + l1a @ Wout[128:256] + bout (4 col tiles)
    for (int t = 0; t < 4; ++t) {
        v8f acc = {};
#pragma unroll
        for (int kc = 0; kc < 32; ++kc)
            acc = wmma_f32(ax[kc], b_frag_g(Wout, 64, kc * 4, t * 16, lane), acc);
#pragma unroll
        for (int kc = 0; kc < 32; ++kc)
            acc = wmma_f32(al[kc], b_frag_g(Wout, 64, 128 + kc * 4, t * 16, lane), acc);
        store_tile(l1_out, 64, r0, t * 16, acc, lane, bout[t * 16 + n]);
    }
}

// ---------------------------------------------------------------------------
// GCN aggregation: g[i] = bias + h[i]*dis[i]^2 (self-loop) then edge scatter
// ---------------------------------------------------------------------------
template <int W>
__global__ void seg_init_kernel(const float* __restrict__ h, const float* __restrict__ dis,
                                const float* __restrict__ bias, float* __restrict__ g, int N)
{
    int idx = blockIdx.x * blockDim.x + threadIdx.x;
    if (idx >= N * W) return;
    int i = idx / W, j = idx % W;
    float d = dis[i];
    g[idx] = bias[j] + h[idx] * d * d;
}

template <int W>
__global__ void seg_scatter_kernel(const int* __restrict__ row, const int* __restrict__ col,
                                   const float* __restrict__ dis, const float* __restrict__ h,
                                   float* __restrict__ g, int E)
{
    constexpr int PER_EDGE = W / 4;           // threads per edge, 4 floats each
    long long t = (long long)blockIdx.x * blockDim.x + threadIdx.x;
    if (t >= (long long)E * PER_EDGE) return;
    int e = (int)(t / PER_EDGE);
    int j = ((int)(t % PER_EDGE)) << 2;
    int r = row[e], c = col[e];
    float nrm = dis[r] * dis[c];
    const float4 hv = *(const float4*)(h + (size_t)r * W + j);
    float* gp = g + (size_t)c * W + j;
    atomicAdd(gp + 0, hv.x * nrm);
    atomicAdd(gp + 1, hv.y * nrm);
    atomicAdd(gp + 2, hv.z * nrm);
    atomicAdd(gp + 3, hv.w * nrm);
}

// ---------------------------------------------------------------------------
// h2 = g1 @ Wg2  (N x 128 -> N x 64), A fragments register-cached
// ---------------------------------------------------------------------------
__global__ __launch_bounds__(128) void gcn2_gemm_kernel(
    const float* __restrict__ g1, const float* __restrict__ Wg2,
    float* __restrict__ h2, int N)
{
    const int lane = threadIdx.x & 31;
    const int wave = threadIdx.x >> 5;
    const int r0   = (blockIdx.x * 4 + wave) * 16;
    if (r0 >= N) return;

    v2f ag[32];
#pragma unroll
    for (int kc = 0; kc < 32; ++kc)
        ag[kc] = a_frag_g(g1, 128, r0, kc * 4, lane);

    for (int t = 0; t < 4; ++t) {
        v8f acc = {};
#pragma unroll
        for (int kc = 0; kc < 32; ++kc)
            acc = wmma_f32(ag[kc], b_frag_g(Wg2, 64, kc * 4, t * 16, lane), acc);
        store_tile(h2, 64, r0, t * 16, acc, lane, 0.f);
    }
}

// ---------------------------------------------------------------------------
// x2 = [x_neighbor, g2] @ W_out + b_out  (concat GEMM: K = 128 + 64)
// ---------------------------------------------------------------------------
__global__ __launch_bounds__(128) void out_gemm_kernel(
    const float* __restrict__ xn, const float* __restrict__ g2,
    const float* __restrict__ W, const float* __restrict__ b,
    float* __restrict__ out, int N)
{
    const int lane = threadIdx.x & 31;
    const int wave = threadIdx.x >> 5;
    const int r0   = (blockIdx.x * 4 + wave) * 16;
    if (r0 >= N) return;
    const int n = lane & 15;

    v2f axn[32];
#pragma unroll
    for (int kc = 0; kc < 32; ++kc)
        axn[kc] = a_frag_g(xn, 128, r0, kc * 4, lane);
    v2f ag2[16];
#pragma unroll
    for (int kc = 0; kc < 16; ++kc)
        ag2[kc] = a_frag_g(g2, 64, r0, kc * 4, lane);

    for (int t = 0; t < 4; ++t) {
        v8f acc = {};
#pragma unroll
        for (int kc = 0; kc < 32; ++kc)
            acc = wmma_f32(axn[kc], b_frag_g(W, 64, kc * 4, t * 16, lane), acc);
#pragma unroll
        for (int kc = 0; kc < 16; ++kc)
            acc = wmma_f32(ag2[kc], b_frag_g(W, 64, 128 + kc * 4, t * 16, lane), acc);
        store_tile(out, 64, r0, t * 16, acc, lane, b[t * 16 + n]);
    }
}

// ---------------------------------------------------------------------------
// Launcher
// ---------------------------------------------------------------------------
extern "C" void kernel_launch(void* const* d_in, const int* in_sizes, int n_in,
                              void* d_out, int out_size, void* d_ws, size_t ws_size,
                              hipStream_t stream) {
    const float* x_self     = (const float*)d_in[0];
    const float* x_neighbor = (const float*)d_in[1];
    const int*   edge_index = (const int*)d_in[2];
    const float* W_in_self  = (const float*)d_in[3];
    const float* b_in_self  = (const float*)d_in[4];
    const float* W_out_self = (const float*)d_in[5];
    const float* b_out_self = (const float*)d_in[6];
    const float* Wg1        = (const float*)d_in[7];
    const float* bg1        = (const float*)d_in[8];
    const float* Wg2        = (const float*)d_in[9];
    const float* bg2        = (const float*)d_in[10];
    const float* W_out      = (const float*)d_in[11];
    const float* b_out      = (const float*)d_in[12];

    const int N = in_sizes[0] / 128;
    const int E = in_sizes[2] / 2;
    const int* row = edge_index;       // edge_index[0]
    const int* col = edge_index + E;   // edge_index[1]

    // Workspace layout (floats): dis[N] | deg[N] | h[N*128] | g[N*128]
    float* ws  = (float*)d_ws;
    float* dis = ws;
    float* deg = ws + N;
    float* h   = ws + 2 * (size_t)N;          // h1, later reused for h2
    float* g   = h + (size_t)N * 128;         // g1, later reused for g2

    float* l1_out = (float*)d_out;
    float* x2_out = (float*)d_out + (size_t)N * 64;

    // 1) degree + normalization
    deg_init_kernel<<<(N + 255) / 256, 256, 0, stream>>>(deg, N);
    deg_acc_kernel<<<(E + 255) / 256, 256, 0, stream>>>(col, deg, E);
    dis_kernel<<<(N + 255) / 256, 256, 0, stream>>>(deg, dis, N);

    const int strips = (N + 15) / 16;          // 2500 (exact)
    const int blocks = (strips + 3) / 4;       // 625 blocks x 4 waves

    // 2) self branch (l1) + h1 = x_self @ Wg1
    self_branch_kernel<<<blocks, 128, 0, stream>>>(x_self, W_in_self, b_in_self,
                                                   W_out_self, b_out_self, Wg1,
                                                   l1_out, h, N);

    // 3) gcn1 aggregation: g1 = bg1 + self-loop + edge scatter
    seg_init_kernel<128><<<((N * 128) + 255) / 256, 256, 0, stream>>>(h, dis, bg1, g, N);
    seg_scatter_kernel<128><<<(int)(((long long)E * 32 + 255) / 256), 256, 0, stream>>>(
        row, col, dis, h, g, E);

    // 4) h2 = g1 @ Wg2 (reuse h buffer)
    gcn2_gemm_kernel<<<blocks, 128, 0, stream>>>(g, Wg2, h, N);

    // 5) gcn2 aggregation: g2 = bg2 + self-loop + edge scatter (reuse g buffer)
    seg_init_kernel<64><<<((N * 64) + 255) / 256, 256, 0, stream>>>(h, dis, bg2, g, N);
    seg_scatter_kernel<64><<<(int)(((long long)E * 16 + 255) / 256), 256, 0, stream>>>(
        row, col, dis, h, g, E);

    // 6) x2 = [x_neighbor, g2] @ W_out + b_out
    out_gemm_kernel<<<blocks, 128, 0, stream>>>(x_neighbor, g, W_out, b_out, x2_out, N);
}